// SpatialWarp_64441689309867
// MI455X (gfx1250) — compile-verified
//
#include <hip/hip_runtime.h>

// SpatialWarp for MI455X (gfx1250): bandwidth-bound bilinear gather.
// B,T,H,W = 8,16,512,512 (fp32). U/out streamed with NT hints (128-bit),
// img (8MB) kept hot in the 192MB L2 for the 4 gathers per output.

#define BB 8
#define TT 16
#define HH 512
#define WW 512

typedef float v4f __attribute__((ext_vector_type(4)));

// 4 outputs (one float4) per thread per iteration, 4 iterations per thread.
// Block of 256 threads (8 wave32) covers 1024 vec4 = 4096 outputs, contiguous.
#define VECS_PER_BLOCK (256 * 4)

__launch_bounds__(256, 8)
__global__ void SpatialWarp_64441689309867_kernel(const float* __restrict__ img,
                                                  const float* __restrict__ U,
                                                  float* __restrict__ out,
                                                  int n_vec) {
    const int vbase = blockIdx.x * VECS_PER_BLOCK + threadIdx.x;

#pragma unroll
    for (int k = 0; k < 4; ++k) {
        const int v = vbase + k * 256;
        if (v >= n_vec) break;

        // Prefetch next iteration's U chunk (gfx1250 global_prefetch_b8).
        // Speculative: OOB prefetches are silently dropped.
        if (k < 3) {
            __builtin_prefetch((const void*)(U + ((size_t)(v + 256) * 8)), 0, 1);
        }

        const size_t n0 = (size_t)v * 4;           // first of 4 consecutive outputs
        const int w0  = (int)(n0 & (WW - 1));      // W = 512 divisible by 4: same row
        const int row = (int)(n0 >> 9);            // (b*T + t)*H + h
        const int h   = row & (HH - 1);
        const int bt  = row >> 9;
        const int b   = bt >> 4;                   // T = 16
        const float* __restrict__ imgb = img + ((size_t)b << 18);  // b*H*W

        // U stream: 8 floats (4 samples x {di,dj}) = two NT b128 loads.
        const v4f u01 = __builtin_nontemporal_load((const v4f*)(U + n0 * 2));
        const v4f u23 = __builtin_nontemporal_load((const v4f*)(U + n0 * 2 + 4));

        const float disp_i[4] = {u01.x, u01.z, u23.x, u23.z};
        const float disp_j[4] = {u01.y, u01.w, u23.y, u23.w};
        const float hf = (float)h;

        v4f res;
#pragma unroll
        for (int s = 0; s < 4; ++s) {
            const float si = hf - disp_i[s];
            const float sj = (float)(w0 + s) - disp_j[s];

            // floor -> int32 -> clip (matches reference ordering exactly)
            int i0 = (int)floorf(si);
            int j0 = (int)floorf(sj);
            i0 = min(max(i0, 0), HH - 1);
            j0 = min(max(j0, 0), WW - 1);
            const int i1 = min(i0 + 1, HH - 1);
            const int j1 = min(j0 + 1, WW - 1);

            // fractional weights computed AFTER clipping
            const float di = si - (float)i0;
            const float dj = sj - (float)j0;

            const float* __restrict__ r0 = imgb + ((size_t)i0 << 9);
            const float* __restrict__ r1 = imgb + ((size_t)i1 << 9);
            const float f00 = r0[j0];
            const float f01 = r0[j1];
            const float f10 = r1[j0];
            const float f11 = r1[j1];

            const float f0 = fmaf(f01 - f00, dj, f00);
            const float f1 = fmaf(f11 - f10, dj, f10);
            res[s] = fmaf(f1 - f0, di, f0);
        }

        // NT b128 store: don't let the 134MB output evict img from L2.
        __builtin_nontemporal_store(res, (v4f*)(out + n0));
    }
}

extern "C" void kernel_launch(void* const* d_in, const int* in_sizes, int n_in,
                              void* d_out, int out_size, void* d_ws, size_t ws_size,
                              hipStream_t stream) {
    const float* img = (const float*)d_in[0];   // (B,H,W)       = 8*512*512
    const float* U   = (const float*)d_in[1];   // (B,T,H,W,2)
    float* out       = (float*)d_out;           // (B,T,H,W)

    const int n_vec  = out_size / 4;            // float4 units
    const int blocks = (n_vec + VECS_PER_BLOCK - 1) / VECS_PER_BLOCK;

    SpatialWarp_64441689309867_kernel<<<blocks, 256, 0, stream>>>(img, U, out, n_vec);
}